// Emb_ATT_Layers_45818711114145
// MI455X (gfx1250) — compile-verified
//
#include <hip/hip_runtime.h>
#include <hip/hip_bf16.h>

#define S_TOK 4
#define DIM   64
#define NHEAD 4
#define HD    16
#define NREL  16
#define HIDD  64
#define LBLD  16

typedef __attribute__((ext_vector_type(2))) float    v2f;
typedef __attribute__((ext_vector_type(8))) float    v8f;
typedef __attribute__((ext_vector_type(4))) unsigned v4u;
typedef __attribute__((ext_vector_type(8))) int      v8i;
typedef __attribute__((ext_vector_type(4))) int      v4i;

// ---------------------------------------------------------------------------
// Generic fp32 WMMA GEMM, K fixed at 64, LDC compile-time.
//   C[M,N] = A[M,64] * B (+bias) (+Cin);  BT: B is W[N,64] row-major (B=W^T)
//   EPI: 0 none, 1 relu, 2 sigmoid
// One wave32 owns one 16-wide N-strip; full B strip (64x16) cached in 16 v2f
// registers; wave grid-strides over M tiles: 16 global_load_b64 + 16 WMMA.
// ---------------------------------------------------------------------------
template <int LDC, bool BT, bool ACC_IN, bool BIAS, int EPI>
__global__ void gemm_f32_wmma(const float* __restrict__ A,
                              const float* __restrict__ B, int ldb,
                              const float* __restrict__ bias,
                              const float* __restrict__ Cin,
                              float* __restrict__ Cout,
                              int M, int N, int wavesPerStrip) {
  const int lane = threadIdx.x & 31;
  const int wid  = blockIdx.x * (blockDim.x >> 5) + (threadIdx.x >> 5);
  const int tilesM = M >> 4, tilesN = N >> 4;
  const int strip = wid / wavesPerStrip;
  const int m0    = wid % wavesPerStrip;
  if (strip >= tilesN) return;
  const int tn   = strip << 4;
  const int half = lane >> 4;          // 0: lanes 0-15, 1: lanes 16-31
  const int sub  = lane & 15;

  v2f bfr[16];
#pragma unroll
  for (int kk = 0; kk < 16; ++kk) {
    const int k = kk * 4 + half * 2;
    if (BT) {
      const float* bp = B + (size_t)(tn + sub) * ldb + k;
      bfr[kk].x = bp[0]; bfr[kk].y = bp[1];
    } else {
      const float* bp = B + (size_t)k * ldb + tn + sub;
      bfr[kk].x = bp[0]; bfr[kk].y = bp[ldb];
    }
  }
  const float bv = BIAS ? bias[tn + sub] : 0.f;

  for (int mt = m0; mt < tilesM; mt += wavesPerStrip) {
    const int tm = mt << 4;
    v8f acc = {};
    if (ACC_IN) {
      const float* cin = Cin + (size_t)(tm + half * 8) * LDC + tn + sub;
#pragma unroll
      for (int r = 0; r < 8; ++r) acc[r] = cin[r * LDC];
    }
    const float* ap = A + (size_t)(tm + sub) * 64 + half * 2;
    __builtin_prefetch(A + (size_t)(tm + (wavesPerStrip << 4) + sub) * 64, 0, 1);
#pragma unroll
    for (int kk = 0; kk < 16; ++kk) {
      v2f a; a.x = ap[kk * 4]; a.y = ap[kk * 4 + 1];
      acc = __builtin_amdgcn_wmma_f32_16x16x4_f32(
          false, a, false, bfr[kk], (short)0, acc, false, false);
    }
    float* cp = Cout + (size_t)(tm + half * 8) * LDC + tn + sub;
#pragma unroll
    for (int r = 0; r < 8; ++r) {
      float v = acc[r] + bv;
      if (EPI == 1) v = fmaxf(v, 0.f);
      if (EPI == 2) v = 1.f / (1.f + __expf(-v));
      cp[r * LDC] = v;
    }
  }
}

// ---------------------------------------------------------------------------
// Tensor Data Mover: issue one 64x64 fp32 tile load (16 KB) into LDS.
// D# per CDNA5 ISA ch.8: group0 {count=1, lds_addr, 57b global_addr, type=2},
// group1 {data_size=4B, tensor dims/strides, tile 64x64}; groups 2/3 zero.
// ---------------------------------------------------------------------------
__device__ __forceinline__ void tdm_load_tile64x64(const float* gsrc,
                                                   unsigned ldsAddr, int M) {
  const unsigned long long ga = (unsigned long long)(const void*)gsrc;
  v4u g0;
  g0[0] = 1u;                                   // count=1, user descriptor
  g0[1] = ldsAddr;                              // lds_addr
  g0[2] = (unsigned)ga;                         // global_addr[31:0]
  g0[3] = (unsigned)((ga >> 32) & 0x01FFFFFFu) | (2u << 30);  // [56:32] | type=2
  v8i g1;
  g1[0] = (int)(2u << 16);                                    // data_size = 4B
  g1[1] = (int)(64u << 16);                                   // tensor_dim0 = 64
  g1[2] = (int)(((unsigned)M & 0xFFFFu) << 16);               // tensor_dim1 lo16
  g1[3] = (int)((((unsigned)M >> 16) & 0xFFFFu) | (64u << 16)); // dim1 hi | tile_dim0
  g1[4] = 64;                                                 // tile_dim1=64
  g1[5] = 64;                                                 // tensor_dim0_stride
  g1[6] = 0; g1[7] = 0;
  v4i z4 = {0, 0, 0, 0};
  v8i z8 = {0, 0, 0, 0, 0, 0, 0, 0};
  __builtin_amdgcn_tensor_load_to_lds(g0, g1, z4, z4, z8, 0);
}

// qkv GEMM with double-buffered TDM A-staging: C[M,192] = A[M,64] @ W^T + b.
// Block = 4 wave32s; wave 0 prefetches the next A block via TDM while all
// waves compute on the current one (TENSORcnt in-order => wait <=1).
__global__ void gemm_qkv_tdm(const float* __restrict__ A,
                             const float* __restrict__ W,   // [192,64]
                             const float* __restrict__ bias,
                             float* __restrict__ C,         // ldc = 192
                             int M, int N, int blocksPerStrip) {
  __shared__ float As[2][64 * 64];
  const int lane = threadIdx.x & 31;
  const int wave = threadIdx.x >> 5;           // 0..3
  const int tilesN = N >> 4;
  const int strip = blockIdx.x / blocksPerStrip;
  const int blk0  = blockIdx.x % blocksPerStrip;
  if (strip >= tilesN) return;
  const int tn   = strip << 4;
  const int half = lane >> 4;
  const int sub  = lane & 15;

  v2f bfr[16];
#pragma unroll
  for (int kk = 0; kk < 16; ++kk) {
    const float* bp = W + (size_t)(tn + sub) * 64 + kk * 4 + half * 2;
    bfr[kk].x = bp[0]; bfr[kk].y = bp[1];
  }
  const float bv = bias[tn + sub];
  const unsigned lds0 = (unsigned)(size_t)(&As[0][0]);  // flat low 32b = LDS offset
  const unsigned lds1 = (unsigned)(size_t)(&As[1][0]);

  const int mBlocks = M >> 6;                  // 64-row blocks
  if (wave == 0 && blk0 < mBlocks)
    tdm_load_tile64x64(A + (size_t)blk0 * 64 * 64, lds0, M);

  int buf = 0;
  for (int mb = blk0; mb < mBlocks; mb += blocksPerStrip) {
    const int mbNext = mb + blocksPerStrip;
    if (wave == 0) {
      if (mbNext < mBlocks) {
        tdm_load_tile64x64(A + (size_t)mbNext * 64 * 64, buf ? lds0 : lds1, M);
        __builtin_amdgcn_s_wait_tensorcnt(1);  // oldest (current buf) done
      } else {
        __builtin_amdgcn_s_wait_tensorcnt(0);
      }
    }
    __syncthreads();

    v8f acc = {};
    const float* ar = &As[buf][(wave * 16 + sub) * 64 + half * 2];
#pragma unroll
    for (int kk = 0; kk < 16; ++kk) {
      v2f a; a.x = ar[kk * 4]; a.y = ar[kk * 4 + 1];
      acc = __builtin_amdgcn_wmma_f32_16x16x4_f32(
          false, a, false, bfr[kk], (short)0, acc, false, false);
    }
    float* cp = C + (size_t)(mb * 64 + wave * 16 + half * 8) * 192 + tn + sub;
#pragma unroll
    for (int r = 0; r < 8; ++r) cp[r * 192] = acc[r] + bv;

    __syncthreads();
    buf ^= 1;
  }
}

// ---------------------------------------------------------------------------
__global__ void attention_kernel(const float* __restrict__ qkv,
                                 float* __restrict__ xo, int N) {
  int n = blockIdx.x * blockDim.x + threadIdx.x;
  if (n >= N) return;
  const size_t row0 = (size_t)n * 192;
  float o[DIM];
#pragma unroll
  for (int h = 0; h < NHEAD; ++h) {
    float sc[S_TOK], m = -1e30f;
#pragma unroll
    for (int s = 0; s < S_TOK; ++s) {
      const float* q = qkv + row0 + h * HD;
      const float* k = qkv + ((size_t)s * N + n) * 192 + DIM + h * HD;
      float d = 0.f;
#pragma unroll
      for (int i = 0; i < HD; ++i) d += q[i] * k[i];
      sc[s] = d * 0.25f;
      m = fmaxf(m, sc[s]);
    }
    float sum = 0.f;
#pragma unroll
    for (int s = 0; s < S_TOK; ++s) { sc[s] = __expf(sc[s] - m); sum += sc[s]; }
    const float inv = 1.f / sum;
#pragma unroll
    for (int i = 0; i < HD; ++i) {
      float acc = 0.f;
#pragma unroll
      for (int s = 0; s < S_TOK; ++s)
        acc += sc[s] * qkv[((size_t)s * N + n) * 192 + 2 * DIM + h * HD + i];
      o[h * HD + i] = acc * inv;
    }
  }
#pragma unroll
  for (int i = 0; i < DIM; ++i) xo[(size_t)n * DIM + i] = o[i];
}

__global__ void count_kernel(const int* __restrict__ ei,
                             const int* __restrict__ et,
                             unsigned* __restrict__ cnt, int E) {
  int e = blockIdx.x * blockDim.x + threadIdx.x;
  if (e >= E) return;
  atomicAdd(&cnt[(size_t)ei[E + e] * NREL + et[e]], 1u);
}

// 8 threads per edge -> coalesced fp32 atomics over 64-float rows.
__global__ void scatter_rel_kernel(const float* __restrict__ x,
                                   const int* __restrict__ ei,
                                   const int* __restrict__ et,
                                   const unsigned* __restrict__ cnt,
                                   float* __restrict__ pre, int E, int rel) {
  int t = blockIdx.x * blockDim.x + threadIdx.x;
  int e = t >> 3, part = t & 7;
  if (e >= E) return;
  if (et[e] != rel) return;
  const int src = ei[e], dst = ei[E + e];
  unsigned c = cnt[(size_t)dst * NREL + rel];
  const float inv = 1.f / (float)(c ? c : 1u);
  const float* xs = x + (size_t)src * DIM + part * 8;
  float* pd = pre + (size_t)dst * DIM + part * 8;
#pragma unroll
  for (int i = 0; i < 8; ++i) atomicAdd(&pd[i], xs[i] * inv);
}

// ---------------------------------------------------------------------------
static inline size_t alignup(size_t x) { return (x + 255) & ~(size_t)255; }

extern "C" void kernel_launch(void* const* d_in, const int* in_sizes, int n_in,
                              void* d_out, int out_size, void* d_ws, size_t ws_size,
                              hipStream_t stream) {
  const float* embedding  = (const float*)d_in[0];
  const int*   edge_index = (const int*)d_in[1];
  const int*   edge_type  = (const int*)d_in[2];
  const float* in_proj_w  = (const float*)d_in[3];
  const float* in_proj_b  = (const float*)d_in[4];
  const float* out_proj_w = (const float*)d_in[5];
  const float* out_proj_b = (const float*)d_in[6];
  const float* W1    = (const float*)d_in[7];
  const float* root1 = (const float*)d_in[8];
  const float* bias1 = (const float*)d_in[9];
  const float* W2    = (const float*)d_in[10];
  const float* root2 = (const float*)d_in[11];
  const float* bias2 = (const float*)d_in[12];
  float* out = (float*)d_out;

  const int N  = in_sizes[0] / (S_TOK * DIM);
  const int E  = in_sizes[1] / 2;
  const int SN = S_TOK * N;

  char* ws = (char*)d_ws;
  const size_t qkvBytes   = alignup((size_t)SN * 192 * sizeof(float));
  const size_t xattnBytes = alignup((size_t)N * DIM * sizeof(float));
  float* qkv    = (float*)(ws);
  float* x_attn = (float*)(ws + qkvBytes);
  float* x      = (float*)(ws + qkvBytes + xattnBytes);
  // recycled inside qkv region after out-proj:
  unsigned* cnt = (unsigned*)(ws);
  float* pre    = (float*)(ws + alignup((size_t)N * NREL * sizeof(unsigned)));
  float* agg1   = (float*)((char*)pre + alignup((size_t)N * DIM * sizeof(float)));
  float* agg2   = (float*)((char*)agg1 + alignup((size_t)N * DIM * sizeof(float)));
  (void)ws_size; (void)n_in; (void)out_size;

  const int BLK = 256, WPB = BLK / 32;
  const int WPS = 1024;                         // waves per 16-col strip
  auto gemmGrid = [&](int Nc) { return (Nc >> 4) * WPS / WPB; };

  // 1) qkv = X @ in_proj_w^T + b  -- double-buffered TDM A-staging
  const int BPS = 128;                          // blocks per strip
  gemm_qkv_tdm<<<(192 >> 4) * BPS, 128, 0, stream>>>(
      embedding, in_proj_w, in_proj_b, qkv, SN, 192, BPS);

  // 2) attention (first token) -> x_attn
  attention_kernel<<<(N + BLK - 1) / BLK, BLK, 0, stream>>>(qkv, x_attn, N);

  // 3) x = x_attn @ out_proj_w^T + b
  gemm_f32_wmma<64, true, false, true, 0><<<gemmGrid(DIM), BLK, 0, stream>>>(
      x_attn, out_proj_w, DIM, out_proj_b, nullptr, x, N, DIM, WPS);

  // 4) per-(dst,rel) counts
  (void)hipMemsetAsync(cnt, 0, (size_t)N * NREL * sizeof(unsigned), stream);
  count_kernel<<<(E + BLK - 1) / BLK, BLK, 0, stream>>>(edge_index, edge_type, cnt, E);

  // 5) layer 1: agg1 = x@root1 + bias1; += mean_r(x)@W1[r]; relu folded into r=15
  gemm_f32_wmma<64, false, false, true, 0><<<gemmGrid(HIDD), BLK, 0, stream>>>(
      x, root1, HIDD, bias1, nullptr, agg1, N, HIDD, WPS);
  for (int r = 0; r < NREL; ++r) {
    (void)hipMemsetAsync(pre, 0, (size_t)N * DIM * sizeof(float), stream);
    scatter_rel_kernel<<<((E * 8) + BLK - 1) / BLK, BLK, 0, stream>>>(
        x, edge_index, edge_type, cnt, pre, E, r);
    const float* Wr = W1 + (size_t)r * DIM * HIDD;
    if (r < NREL - 1)
      gemm_f32_wmma<64, false, true, false, 0><<<gemmGrid(HIDD), BLK, 0, stream>>>(
          pre, Wr, HIDD, nullptr, agg1, agg1, N, HIDD, WPS);
    else
      gemm_f32_wmma<64, false, true, false, 1><<<gemmGrid(HIDD), BLK, 0, stream>>>(
          pre, Wr, HIDD, nullptr, agg1, agg1, N, HIDD, WPS);
  }

  // 6) layer 2: agg2 = h1@root2 + bias2; += mean_r(h1)@W2[r]; sigmoid->d_out on r=15
  gemm_f32_wmma<16, false, false, true, 0><<<gemmGrid(LBLD), BLK, 0, stream>>>(
      agg1, root2, LBLD, bias2, nullptr, agg2, N, LBLD, WPS);
  for (int r = 0; r < NREL; ++r) {
    (void)hipMemsetAsync(pre, 0, (size_t)N * DIM * sizeof(float), stream);
    scatter_rel_kernel<<<((E * 8) + BLK - 1) / BLK, BLK, 0, stream>>>(
        agg1, edge_index, edge_type, cnt, pre, E, r);
    const float* Wr = W2 + (size_t)r * HIDD * LBLD;
    if (r < NREL - 1)
      gemm_f32_wmma<16, false, true, false, 0><<<gemmGrid(LBLD), BLK, 0, stream>>>(
          pre, Wr, LBLD, nullptr, agg2, agg2, N, LBLD, WPS);
    else
      gemm_f32_wmma<16, false, true, false, 2><<<gemmGrid(LBLD), BLK, 0, stream>>>(
          pre, Wr, LBLD, nullptr, agg2, out, N, LBLD, WPS);
  }
}